// TransformerBlock_3D_single_deform_LKA_6983616823916
// MI455X (gfx1250) — compile-verified
//
#include <hip/hip_runtime.h>
#include <hip/hip_bf16.h>

// Problem constants (B=2, C=64, 16x16x16 spatial)
constexpr int CH  = 64;
constexpr int NS  = 4096;   // 16^3 spatial positions
constexpr int NB  = 2;      // batch
constexpr int KK  = 27;     // 3x3x3 taps

typedef __attribute__((ext_vector_type(2))) float v2f;
typedef __attribute__((ext_vector_type(8))) float v8f;

// fp32 WMMA: D(16x16,f32) = A(16x4,f32) * B(4x16,f32) + C
// Per-lane layout (wave32): col = lane&15, hi = lane>>4
//   A: M = col, K = {kA, kA+1} with kA = k0 + 2*hi
//   B: N = col, K = {kA, kA+1}
//   C/D: N = col, M = v + 8*hi for vector element v
static __device__ __forceinline__ v8f wmma4(float a0, float a1, float b0, float b1, v8f c) {
  v2f a = {a0, a1};
  v2f b = {b0, b1};
  return __builtin_amdgcn_wmma_f32_16x16x4_f32(false, a, false, b, (short)0, c, false, false);
}

// ---------------- weight repack: W (M x K row-major) -> Wp float2[(K/2) x M] ----
// Wp[k2*M + o] = { W[o*K + 2*k2], W[o*K + 2*k2 + 1] }  (coalesced A-fragment loads)
__global__ void packw_kernel(const float* __restrict__ W, float2* __restrict__ Wp,
                             int M, int K) {
  int i = blockIdx.x * blockDim.x + threadIdx.x;
  if (i >= (K / 2) * M) return;
  int o = i % M, k2 = i / M;
  Wp[i] = make_float2(W[(size_t)o * K + 2 * k2], W[(size_t)o * K + 2 * k2 + 1]);
}

// conv weights W (OC x 64 x 27) -> Wp float2[27][32][OCP], zero-padded rows >= OC
// Wp[(t*32 + k2)*OCP + o] = { W[o][2k2][t], W[o][2k2+1][t] }
__global__ void packconv_kernel(const float* __restrict__ W, float2* __restrict__ Wp,
                                int OC, int OCP) {
  int i = blockIdx.x * blockDim.x + threadIdx.x;
  if (i >= 27 * 32 * OCP) return;
  int o = i % OCP;
  int k2 = (i / OCP) % 32;
  int t = i / (OCP * 32);
  float a = 0.f, b = 0.f;
  if (o < OC) {
    a = W[((size_t)o * 64 + 2 * k2) * 27 + t];
    b = W[((size_t)o * 64 + 2 * k2 + 1) * 27 + t];
  }
  Wp[i] = make_float2(a, b);
}

// ---------------- LayerNorm over channels (tokens = NCDHW -> (B,N,C)) ----------
__global__ void ln_kernel(const float* __restrict__ x, const float* __restrict__ g,
                          const float* __restrict__ bb, float* __restrict__ y) {
  int i = blockIdx.x * blockDim.x + threadIdx.x;  // over B*NS
  if (i >= NB * NS) return;
  int b = i / NS, n = i % NS;
  const float* xb = x + (size_t)b * CH * NS + n;
  float s = 0.f, q = 0.f;
  for (int c = 0; c < CH; ++c) { float v = xb[(size_t)c * NS]; s += v; q += v * v; }
  float mu = s * (1.f / CH);
  float var = q * (1.f / CH) - mu * mu;
  float r = rsqrtf(var + 1e-5f);
  float* yb = y + (size_t)b * CH * NS + n;
  for (int c = 0; c < CH; ++c) {
    float v = xb[(size_t)c * NS];
    yb[(size_t)c * NS] = (v - mu) * r * g[c] + bb[c];
  }
}

// ---------------- Generic M=64 x N=4096 WMMA GEMM with fusions ------------------
// out[o][n] = act( sum_k W[o][k] * in(k,n) + bias[o] ) (+ res[o][n])
// in(k,n) = in0[k*NS+n] * (MUL ? in1[k*NS+n] : 1); Wp is pair-packed K-major.
template <int MUL>
__global__ void gemm_wmma_kernel(const float2* __restrict__ Wp, int K,
                                 const float* __restrict__ bias,
                                 const float* __restrict__ in0,
                                 const float* __restrict__ in1,
                                 const float* __restrict__ res,
                                 float* __restrict__ out, int gelu) {
  int wave = (blockIdx.x * blockDim.x + threadIdx.x) >> 5;  // 256 waves total
  int lane = threadIdx.x & 31;
  int col = lane & 15, hi = lane >> 4;
  int n0 = wave * 16;
  if (n0 >= NS) return;
  v8f acc[4];
#pragma unroll
  for (int m = 0; m < 4; ++m) acc[m] = (v8f){0.f,0.f,0.f,0.f,0.f,0.f,0.f,0.f};

  const float* p0base = in0 + n0 + col;
  const float* p1base = MUL ? (in1 + n0 + col) : nullptr;
  for (int k0 = 0; k0 < K; k0 += 4) {
    int kA = k0 + 2 * hi;
    float b0 = p0base[(size_t)kA * NS];
    float b1 = p0base[(size_t)(kA + 1) * NS];
    if (MUL) {
      b0 *= p1base[(size_t)kA * NS];
      b1 *= p1base[(size_t)(kA + 1) * NS];
    }
    const float2* wrow = Wp + (size_t)(kA >> 1) * 64;
#pragma unroll
    for (int m = 0; m < 4; ++m) {
      float2 a = wrow[m * 16 + col];
      acc[m] = wmma4(a.x, a.y, b0, b1, acc[m]);
    }
  }
#pragma unroll
  for (int m = 0; m < 4; ++m) {
#pragma unroll
    for (int v = 0; v < 8; ++v) {
      int o = m * 16 + v + 8 * hi;
      float val = acc[m][v] + bias[o];
      if (gelu) val = 0.5f * val * (1.f + erff(val * 0.70710678118654752f));
      if (res) val += res[(size_t)o * NS + n0 + col];
      out[(size_t)o * NS + n0 + col] = val;
    }
  }
}

// ---------------- 3x3x3 conv (64 -> OC) via WMMA, pad=1 -------------------------
template <int MT, int OC>
__global__ void conv3_wmma_kernel(const float* __restrict__ in,    // (B,64,NS)
                                  const float2* __restrict__ Wp,   // [27][32][MT*16]
                                  const float* __restrict__ bias,
                                  float* __restrict__ out) {       // (B,OC,NS)
  constexpr int OCP = MT * 16;
  int wave = (blockIdx.x * blockDim.x + threadIdx.x) >> 5;  // NB*256 waves
  int lane = threadIdx.x & 31;
  int col = lane & 15, hi = lane >> 4;
  int b = wave >> 8;
  int n0 = (wave & 255) * 16;
  int h = n0 >> 8, w = (n0 >> 4) & 15, d = col;  // n0 % 16 == 0 -> d = col
  const float* inb = in + (size_t)b * CH * NS;
  v8f acc[MT];
#pragma unroll
  for (int m = 0; m < MT; ++m) acc[m] = (v8f){0.f,0.f,0.f,0.f,0.f,0.f,0.f,0.f};

  for (int t = 0; t < 27; ++t) {
    int kz = t / 9, ky = (t / 3) % 3, kx = t % 3;
    int hs = h + kz - 1, ws = w + ky - 1, ds = d + kx - 1;
    bool valid = ((unsigned)hs < 16u) & ((unsigned)ws < 16u) & ((unsigned)ds < 16u);
    int nsrc = valid ? ((hs * 16 + ws) * 16 + ds) : 0;
    for (int k0 = 0; k0 < CH; k0 += 4) {
      int kA = k0 + 2 * hi;
      float b0 = valid ? inb[(size_t)kA * NS + nsrc] : 0.f;
      float b1 = valid ? inb[(size_t)(kA + 1) * NS + nsrc] : 0.f;
      const float2* wrow = Wp + ((size_t)t * 32 + (kA >> 1)) * OCP;
#pragma unroll
      for (int m = 0; m < MT; ++m) {
        float2 a = wrow[m * 16 + col];
        acc[m] = wmma4(a.x, a.y, b0, b1, acc[m]);
      }
    }
  }
#pragma unroll
  for (int m = 0; m < MT; ++m) {
#pragma unroll
    for (int v = 0; v < 8; ++v) {
      int o = m * 16 + v + 8 * hi;
      if (o < OC)
        out[((size_t)b * OC + o) * NS + n0 + col] = acc[m][v] + bias[o];
    }
  }
}

// ---------------- depthwise conv (groups = C) -----------------------------------
__global__ void dw_kernel(const float* __restrict__ in, const float* __restrict__ w,
                          const float* __restrict__ bias, float* __restrict__ out,
                          int S, int dil, int pad) {
  int i = blockIdx.x * blockDim.x + threadIdx.x;
  if (i >= NB * CH * NS) return;
  int n = i & (NS - 1);
  int c = (i >> 12) & 63;
  int b = i >> 18;
  int h = n >> 8, ww = (n >> 4) & 15, d = n & 15;
  const float* inb = in + ((size_t)b * CH + c) * NS;
  const float* wc = w + (size_t)c * S * S * S;
  float s = bias[c];
  for (int kz = 0; kz < S; ++kz) {
    int hs = h + kz * dil - pad;
    if ((unsigned)hs >= 16u) continue;
    for (int ky = 0; ky < S; ++ky) {
      int ws = ww + ky * dil - pad;
      if ((unsigned)ws >= 16u) continue;
      for (int kx = 0; kx < S; ++kx) {
        int ds = d + kx * dil - pad;
        if ((unsigned)ds >= 16u) continue;
        s += wc[(kz * S + ky) * S + kx] * inb[(hs * 16 + ws) * 16 + ds];
      }
    }
  }
  out[i] = s;
}

// ---------------- deform: offsets -> 8 trilinear (weight, index) per tap --------
__global__ void defprep_kernel(const float* __restrict__ off, float* __restrict__ wgt,
                               int* __restrict__ idx) {
  int i = blockIdx.x * blockDim.x + threadIdx.x;  // B*27*NS
  if (i >= NB * KK * NS) return;
  int n = i % NS;
  int kk = (i / NS) % KK;
  int b = i / (NS * KK);
  int h = n >> 8, w = (n >> 4) & 15, d = n & 15;
  const float* ob = off + (size_t)b * (3 * KK) * NS;
  float z  = (float)(h + (kk / 9) - 1)       + ob[(size_t)(kk * 3 + 0) * NS + n];
  float yq = (float)(w + ((kk / 3) % 3) - 1) + ob[(size_t)(kk * 3 + 1) * NS + n];
  float xq = (float)(d + (kk % 3) - 1)       + ob[(size_t)(kk * 3 + 2) * NS + n];
  float z0 = floorf(z), y0 = floorf(yq), x0 = floorf(xq);
#pragma unroll
  for (int t = 0; t < 8; ++t) {
    float zi = z0 + (float)(t >> 2);
    float yi = y0 + (float)((t >> 1) & 1);
    float xi = x0 + (float)(t & 1);
    float wv = (1.f - fabsf(z - zi)) * (1.f - fabsf(yq - yi)) * (1.f - fabsf(xq - xi));
    bool valid = (zi >= 0.f) & (zi < 16.f) & (yi >= 0.f) & (yi < 16.f) & (xi >= 0.f) & (xi < 16.f);
    wv = valid ? wv : 0.f;
    int zc = (int)fminf(fmaxf(zi, 0.f), 15.f);
    int yc = (int)fminf(fmaxf(yi, 0.f), 15.f);
    int xc = (int)fminf(fmaxf(xi, 0.f), 15.f);
    size_t o = ((size_t)(b * KK + kk) * 8 + t) * NS + n;
    wgt[o] = wv;
    idx[o] = (zc * 16 + yc) * 16 + xc;
  }
}

// ---------------- deform: gather-sample val[(c*27+kk)][n] for batch b -----------
__global__ void defsample_kernel(const float* __restrict__ a2, const float* __restrict__ wgt,
                                 const int* __restrict__ idx, float* __restrict__ val, int b) {
  int i = blockIdx.x * blockDim.x + threadIdx.x;  // 27*NS
  if (i >= KK * NS) return;
  int n = i % NS;
  int kk = i / NS;
  float wv[8];
  int id[8];
#pragma unroll
  for (int t = 0; t < 8; ++t) {
    size_t o = ((size_t)(b * KK + kk) * 8 + t) * NS + n;
    wv[t] = wgt[o];
    id[t] = idx[o];
  }
  const float* xb = a2 + (size_t)b * CH * NS;
  for (int c = 0; c < CH; ++c) {
    const float* xr = xb + (size_t)c * NS;
    float s = 0.f;
#pragma unroll
    for (int t = 0; t < 8; ++t) s += wv[t] * xr[id[t]];
    val[((size_t)c * KK + kk) * NS + n] = s;
  }
}

// ---------------- batchnorm statistics (per channel over B*N) -------------------
__global__ void bnstats_kernel(const float* __restrict__ x, float* __restrict__ mu,
                               float* __restrict__ var) {
  int c = blockIdx.x;  // 64 blocks
  __shared__ float ss[256], sq[256];
  float s = 0.f, q = 0.f;
  for (int i = threadIdx.x; i < NB * NS; i += blockDim.x) {
    int b = i >> 12, n = i & (NS - 1);
    float v = x[((size_t)b * CH + c) * NS + n];
    s += v;
    q += v * v;
  }
  ss[threadIdx.x] = s;
  sq[threadIdx.x] = q;
  __syncthreads();
  for (int st = 128; st > 0; st >>= 1) {
    if ((int)threadIdx.x < st) {
      ss[threadIdx.x] += ss[threadIdx.x + st];
      sq[threadIdx.x] += sq[threadIdx.x + st];
    }
    __syncthreads();
  }
  if (threadIdx.x == 0) {
    float m = ss[0] * (1.f / (NB * NS));
    mu[c] = m;
    var[c] = sq[0] * (1.f / (NB * NS)) - m * m;
  }
}

__global__ void bnapply_kernel(const float* __restrict__ x, const float* __restrict__ mu,
                               const float* __restrict__ var, const float* __restrict__ g,
                               const float* __restrict__ bb, const float* __restrict__ add,
                               float* __restrict__ out, int leaky) {
  int i = blockIdx.x * blockDim.x + threadIdx.x;
  if (i >= NB * CH * NS) return;
  int c = (i >> 12) & 63;
  float v = (x[i] - mu[c]) * rsqrtf(var[c] + 1e-5f) * g[c] + bb[c];
  if (add) v += add[i];
  if (leaky) v = v > 0.f ? v : 0.01f * v;
  out[i] = v;
}

// skip = x + gamma[c] * y2   (transpose pair cancels; stays NCDHW)
__global__ void skip_kernel(const float* __restrict__ x, const float* __restrict__ gamma,
                            const float* __restrict__ y2, float* __restrict__ skip) {
  int i = blockIdx.x * blockDim.x + threadIdx.x;
  if (i >= NB * CH * NS) return;
  int c = (i >> 12) & 63;
  skip[i] = x[i] + gamma[c] * y2[i];
}

extern "C" void kernel_launch(void* const* d_in, const int* in_sizes, int n_in,
                              void* d_out, int out_size, void* d_ws, size_t ws_size,
                              hipStream_t stream) {
  const float* x     = (const float*)d_in[0];
  const float* ln_g  = (const float*)d_in[1];
  const float* ln_b  = (const float*)d_in[2];
  const float* gamma = (const float*)d_in[3];
  const float* p1_w  = (const float*)d_in[4];
  const float* p1_b  = (const float*)d_in[5];
  const float* c0_w  = (const float*)d_in[6];
  const float* c0_b  = (const float*)d_in[7];
  const float* cs_w  = (const float*)d_in[8];
  const float* cs_b  = (const float*)d_in[9];
  const float* off_w = (const float*)d_in[10];
  const float* off_b = (const float*)d_in[11];
  const float* def_w = (const float*)d_in[12];
  const float* def_b = (const float*)d_in[13];
  const float* lk1_w = (const float*)d_in[14];
  const float* lk1_b = (const float*)d_in[15];
  const float* p2_w  = (const float*)d_in[16];
  const float* p2_b  = (const float*)d_in[17];
  const float* r1_w  = (const float*)d_in[18];
  const float* r1_b  = (const float*)d_in[19];
  const float* bn1_g = (const float*)d_in[20];
  const float* bn1_b = (const float*)d_in[21];
  const float* r2_w  = (const float*)d_in[22];
  const float* r2_b  = (const float*)d_in[23];
  const float* bn2_g = (const float*)d_in[24];
  const float* bn2_b = (const float*)d_in[25];
  const float* c8_w  = (const float*)d_in[26];
  const float* c8_b  = (const float*)d_in[27];
  float* out = (float*)d_out;

  // Workspace layout (floats). Total ~15M floats (~60 MB).
  const size_t T = (size_t)NB * CH * NS;  // 524288
  float* ws = (float*)d_ws;
  float* y    = ws;            // LN output (shortcut)
  float* u    = ws + T;        // gelu(p1)
  float* t0   = ws + 2 * T;    // a (c0) -> a3 -> o1
  float* t1   = ws + 3 * T;    // a2 (cs) -> y2
  float* t2   = ws + 4 * T;    // defout / skip
  float* t3   = ws + 5 * T;    // r1out -> o2
  float* offb = ws + 6 * T;                       // (B,81,NS) = 663552
  float* wgt  = offb + (size_t)NB * 81 * NS;      // (B,27,8,NS) = 1769472
  int*   idx  = (int*)(wgt + (size_t)NB * KK * 8 * NS);  // same count, int
  float* val  = (float*)idx + (size_t)NB * KK * 8 * NS;  // (1728,NS) per batch, reused
  float* stats = val + (size_t)CH * KK * NS;      // mu1,var1,mu2,var2 (4*64)
  float* mu1 = stats, *var1 = stats + 64, *mu2 = stats + 128, *var2 = stats + 192;
  // pair-packed weights (float2 counts): 1x1 = 2048 each; def = 55296;
  // off = 82944; r1/r2 = 55296 each
  float2* pk_p1  = (float2*)(stats + 256);
  float2* pk_lk1 = pk_p1 + 2048;
  float2* pk_p2  = pk_lk1 + 2048;
  float2* pk_c8  = pk_p2 + 2048;
  float2* pk_def = pk_c8 + 2048;
  float2* pk_off = pk_def + 55296;
  float2* pk_r1  = pk_off + 82944;
  float2* pk_r2  = pk_r1 + 55296;

  const int EW = 256;                 // elementwise block
  const int EB = (int)(T / EW);       // 2048 blocks for B*C*N
  const int GB = 32;                  // 256 waves for M=64 x N=4096 GEMM
  const int CB = 64;                  // 512 waves for conv3 (both batches)

  // 0) repack weights (coalesced A-fragment layout)
  packw_kernel<<<8, EW, 0, stream>>>(p1_w, pk_p1, 64, 64);
  packw_kernel<<<8, EW, 0, stream>>>(lk1_w, pk_lk1, 64, 64);
  packw_kernel<<<8, EW, 0, stream>>>(p2_w, pk_p2, 64, 64);
  packw_kernel<<<8, EW, 0, stream>>>(c8_w, pk_c8, 64, 64);
  packw_kernel<<<216, EW, 0, stream>>>(def_w, pk_def, 64, 1728);
  packconv_kernel<<<324, EW, 0, stream>>>(off_w, pk_off, 81, 96);
  packconv_kernel<<<216, EW, 0, stream>>>(r1_w, pk_r1, 64, 64);
  packconv_kernel<<<216, EW, 0, stream>>>(r2_w, pk_r2, 64, 64);

  // 1) LayerNorm
  ln_kernel<<<(NB * NS) / EW, EW, 0, stream>>>(x, ln_g, ln_b, y);

  // 2) u = gelu(p1 @ y + b)
  for (int b = 0; b < NB; ++b)
    gemm_wmma_kernel<0><<<GB, 256, 0, stream>>>(pk_p1, CH, p1_b,
                                                y + (size_t)b * CH * NS, nullptr, nullptr,
                                                u + (size_t)b * CH * NS, 1);
  // 3) a = dwconv5(u), pad2
  dw_kernel<<<EB, EW, 0, stream>>>(u, c0_w, c0_b, t0, 5, 1, 2);
  // 4) a2 = dwconv7 dil3 (a), pad9
  dw_kernel<<<EB, EW, 0, stream>>>(t0, cs_w, cs_b, t1, 7, 3, 9);

  // 5) offsets = conv3x3(a2) -> 81 channels
  conv3_wmma_kernel<6, 81><<<CB, 256, 0, stream>>>(t1, pk_off, off_b, offb);

  // 6) trilinear weights + indices
  defprep_kernel<<<(NB * KK * NS + EW - 1) / EW, EW, 0, stream>>>(offb, wgt, idx);

  // 7) per batch: sample val, deform GEMM (K=1728)
  for (int b = 0; b < NB; ++b) {
    defsample_kernel<<<(KK * NS + EW - 1) / EW, EW, 0, stream>>>(t1, wgt, idx, val, b);
    gemm_wmma_kernel<0><<<GB, 256, 0, stream>>>(pk_def, CH * KK, def_b,
                                                val, nullptr, nullptr,
                                                t2 + (size_t)b * CH * NS, 0);
  }

  // 8) a3 = lk1 @ defout + b
  for (int b = 0; b < NB; ++b)
    gemm_wmma_kernel<0><<<GB, 256, 0, stream>>>(pk_lk1, CH, lk1_b,
                                                t2 + (size_t)b * CH * NS, nullptr, nullptr,
                                                t0 + (size_t)b * CH * NS, 0);
  // 9) y2 = p2 @ (u * a3) + b + shortcut(y)
  for (int b = 0; b < NB; ++b)
    gemm_wmma_kernel<1><<<GB, 256, 0, stream>>>(pk_p2, CH, p2_b,
                                                u + (size_t)b * CH * NS,
                                                t0 + (size_t)b * CH * NS,
                                                y + (size_t)b * CH * NS,
                                                t1 + (size_t)b * CH * NS, 0);
  // 10) skip = x + gamma*y2
  skip_kernel<<<EB, EW, 0, stream>>>(x, gamma, t1, t2);

  // 11-13) r1 conv -> bn1 -> leaky
  conv3_wmma_kernel<4, 64><<<CB, 256, 0, stream>>>(t2, pk_r1, r1_b, t3);
  bnstats_kernel<<<64, 256, 0, stream>>>(t3, mu1, var1);
  bnapply_kernel<<<EB, EW, 0, stream>>>(t3, mu1, var1, bn1_g, bn1_b, nullptr, t0, 1);

  // 14-16) r2 conv -> bn2 -> +skip -> leaky
  conv3_wmma_kernel<4, 64><<<CB, 256, 0, stream>>>(t0, pk_r2, r2_b, u);
  bnstats_kernel<<<64, 256, 0, stream>>>(u, mu2, var2);
  bnapply_kernel<<<EB, EW, 0, stream>>>(u, mu2, var2, bn2_g, bn2_b, t2, t3, 1);

  // 17) out = skip + c8 @ o2 + b
  for (int b = 0; b < NB; ++b)
    gemm_wmma_kernel<0><<<GB, 256, 0, stream>>>(pk_c8, CH, c8_b,
                                                t3 + (size_t)b * CH * NS, nullptr,
                                                t2 + (size_t)b * CH * NS,
                                                out + (size_t)b * CH * NS, 0);
}